// PairwiseScore_60421599920629
// MI455X (gfx1250) — compile-verified
//
#include <hip/hip_runtime.h>
#include <hip/hip_bf16.h>
#include <math.h>

// ---------------------------------------------------------------------------
// PairwiseScore: out[b,j,k] = sum_m v[m] * tanh( (X W1^T)[b,j,m] + (X W2^T)[b,k,m] )
// B=8, D=256, DIM=128.  Stage 1: fp32 WMMA GEMMs -> workspace.  Stage 2: tiled
// pairwise tanh-reduce using hardware v_tanh_f32.
// ---------------------------------------------------------------------------

#define BATCH 8
#define DNODES 256
#define DIMF 128
#define ROWS (BATCH * DNODES)   // 2048 rows of the flattened [B*D, DIM] matrices

typedef __attribute__((ext_vector_type(2))) float v2f;
typedef __attribute__((ext_vector_type(8))) float v8f;

#if __has_builtin(__builtin_amdgcn_tanhf)
__device__ __forceinline__ float fast_tanh(float x) { return __builtin_amdgcn_tanhf(x); }
#else
__device__ __forceinline__ float fast_tanh(float x) { return tanhf(x); }
#endif

// ---------------------------------------------------------------------------
// Stage 1: A = X @ W1^T and C = X @ W2^T via V_WMMA_F32_16X16X4_F32.
// One wave per 16x16 output tile; both GEMMs share the X (A-matrix) operand.
// Grid: 1024 tiles total (128 row-tiles x 8 col-tiles), 8 waves / block.
// ---------------------------------------------------------------------------
__global__ __launch_bounds__(256) void pairwise_gemm_wmma(
    const float* __restrict__ X,   // [2048, 128]
    const float* __restrict__ W1,  // [128, 128]
    const float* __restrict__ W2,  // [128, 128]
    float* __restrict__ Aout,      // [2048, 128]
    float* __restrict__ Cout)      // [2048, 128]
{
    const int lane  = threadIdx.x & 31;
    const int wave  = threadIdx.x >> 5;
    const int tile  = blockIdx.x * 8 + wave;   // 0..1023
    const int rowt  = tile >> 3;               // 0..127
    const int colt  = tile & 7;                // 0..7

    // A-matrix 16x4 f32 layout: lanes 0-15 row M=lane, K={k0,k0+1};
    // lanes 16-31 row M=lane-16, K={k0+2,k0+3}.
    const int r    = lane & 15;
    const int koff = (lane >> 4) << 1;

    const float* __restrict__ xrow  = X  + (size_t)(rowt * 16 + r) * DIMF;
    // B-matrix 4x16 (K x N), B[k][n] = W[n][k]: column n = lane&15, same K split.
    const float* __restrict__ w1row = W1 + (size_t)(colt * 16 + r) * DIMF;
    const float* __restrict__ w2row = W2 + (size_t)(colt * 16 + r) * DIMF;

    v8f acc1 = {};
    v8f acc2 = {};

#pragma unroll
    for (int k0 = 0; k0 < DIMF; k0 += 4) {
        v2f a  = *(const v2f*)(xrow  + k0 + koff);
        v2f b1 = *(const v2f*)(w1row + k0 + koff);
        v2f b2 = *(const v2f*)(w2row + k0 + koff);
        acc1 = __builtin_amdgcn_wmma_f32_16x16x4_f32(
            /*neg_a=*/false, a, /*neg_b=*/false, b1,
            /*c_mod=*/(short)0, acc1, /*reuse_a=*/false, /*reuse_b=*/false);
        acc2 = __builtin_amdgcn_wmma_f32_16x16x4_f32(
            /*neg_a=*/false, a, /*neg_b=*/false, b2,
            /*c_mod=*/(short)0, acc2, /*reuse_a=*/false, /*reuse_b=*/false);
    }

    // C/D 16x16 f32 layout: VGPR i -> row i + 8*(lane>=16), col lane&15.
    const int mbase = rowt * 16 + ((lane >> 4) << 3);
    const int col   = colt * 16 + (lane & 15);
#pragma unroll
    for (int i = 0; i < 8; ++i) {
        Aout[(size_t)(mbase + i) * DIMF + col] = acc1[i];
        Cout[(size_t)(mbase + i) * DIMF + col] = acc2[i];
    }
}

// ---------------------------------------------------------------------------
// Stage 2: out[b,j,k] = sum_m v[m] * tanh(A[b,j,m] + C[b,k,m]).
// Block = 256 threads -> one 16x16 (j,k) tile per block; A/C/v staged in LDS.
// Row stride 132 floats: 16B-aligned rows (ds_load_b128) and bank-conflict-free
// (tk*132 mod 64 banks = tk*4 -> 16 distinct banks across a half-wave).
// ---------------------------------------------------------------------------
#define PW_STRIDE 132

__global__ __launch_bounds__(256) void pairwise_tanh_reduce(
    const float* __restrict__ A,   // [B*D, 128]
    const float* __restrict__ C,   // [B*D, 128]
    const float* __restrict__ v,   // [128]
    float* __restrict__ out)       // [B, D, D]
{
    __shared__ __align__(16) float sA[16 * PW_STRIDE];
    __shared__ __align__(16) float sC[16 * PW_STRIDE];
    __shared__ __align__(16) float sV[DIMF];

    const int kt  = blockIdx.x;   // 0..15
    const int jt  = blockIdx.y;   // 0..15
    const int b   = blockIdx.z;   // 0..7
    const int tid = threadIdx.x;  // 0..255

    const float4* __restrict__ Abase = (const float4*)(A + (size_t)(b * DNODES + jt * 16) * DIMF);
    const float4* __restrict__ Cbase = (const float4*)(C + (size_t)(b * DNODES + kt * 16) * DIMF);

    // 16 rows x 32 float4 = 512 float4 per tile; 256 threads -> 2 each.
#pragma unroll
    for (int i = tid; i < 16 * (DIMF / 4); i += 256) {
        const int rr = i >> 5;          // row 0..15
        const int cc = i & 31;          // float4 col 0..31
        *(float4*)(&sA[rr * PW_STRIDE + cc * 4]) = Abase[rr * (DIMF / 4) + cc];
        *(float4*)(&sC[rr * PW_STRIDE + cc * 4]) = Cbase[rr * (DIMF / 4) + cc];
    }
    if (tid < DIMF) sV[tid] = v[tid];
    __syncthreads();

    const int tk = tid & 15;
    const int tj = tid >> 4;
    const float* __restrict__ arow = &sA[tj * PW_STRIDE];
    const float* __restrict__ crow = &sC[tk * PW_STRIDE];

    float sum = 0.0f;
#pragma unroll
    for (int m = 0; m < DIMF; m += 4) {
        float4 a4 = *(const float4*)(arow + m);
        float4 c4 = *(const float4*)(crow + m);
        float4 v4 = *(const float4*)(sV + m);
        sum += v4.x * fast_tanh(a4.x + c4.x);
        sum += v4.y * fast_tanh(a4.y + c4.y);
        sum += v4.z * fast_tanh(a4.z + c4.z);
        sum += v4.w * fast_tanh(a4.w + c4.w);
    }

    out[(size_t)b * DNODES * DNODES + (size_t)(jt * 16 + tj) * DNODES + (kt * 16 + tk)] = sum;
}

extern "C" void kernel_launch(void* const* d_in, const int* in_sizes, int n_in,
                              void* d_out, int out_size, void* d_ws, size_t ws_size,
                              hipStream_t stream) {
    const float* X  = (const float*)d_in[0];  // [8,256,128]
    const float* W1 = (const float*)d_in[1];  // [128,128]
    const float* W2 = (const float*)d_in[2];  // [128,128]
    const float* v  = (const float*)d_in[3];  // [128]
    float* out = (float*)d_out;               // [8,256,256]

    float* A = (float*)d_ws;                          // 2048*128 floats = 1 MB
    float* C = A + (size_t)ROWS * DIMF;               // next 1 MB

    // Stage 1: 1024 wave-tiles, 8 waves per 256-thread block -> 128 blocks.
    pairwise_gemm_wmma<<<128, 256, 0, stream>>>(X, W1, W2, A, C);

    // Stage 2: one 16x16 (j,k) tile per block.
    dim3 grid(DNODES / 16, DNODES / 16, BATCH);
    pairwise_tanh_reduce<<<grid, 256, 0, stream>>>(A, C, v, out);
}